// TransE_32710470926683
// MI455X (gfx1250) — compile-verified
//
#include <hip/hip_runtime.h>
#include <hip/hip_bf16.h>

// CDNA5 / gfx1250: wave32, WMMA F32_16X16X4_F32 for the E x 8 = (E x D) * (D x B) GEMM
// inside score[b,e] = sqrt(||t_e||^2 - 2 t_e.p_b + ||p_b||^2) + bias_head[b] + bias_tail[e]

typedef __attribute__((ext_vector_type(2))) float v2f;
typedef __attribute__((ext_vector_type(8))) float v8f;

#define EMB_DIM 128
#define BATCH   8

__global__ __launch_bounds__(256)
void transe_score_wmma(const int*   __restrict__ head,
                       const int*   __restrict__ relation,
                       const int*   __restrict__ tail,
                       const float* __restrict__ emb_entity,
                       const float* __restrict__ emb_relation,
                       const float* __restrict__ bias_head,
                       const float* __restrict__ bias_tail,
                       float*       __restrict__ out,
                       int E)
{
    // pred (query) matrix staged in LDS, padded to 16 rows (rows 8..15 zero)
    __shared__ float sPred[16 * EMB_DIM];   // 8 KB
    __shared__ float sQ[16];                // ||p_b||^2
    __shared__ float sBH[16];               // bias_head gathered per query
    __shared__ float sTn[8 * 16];           // per-wave ||t||^2 for its 16 rows

    const int tid = threadIdx.x;

    // ---- stage pred[b][d] = emb_entity[head[b]][d] + emb_relation[relation[b]][d]
    for (int idx = tid; idx < 16 * EMB_DIM; idx += 256) {
        const int b = idx >> 7;      // row (query)
        const int d = idx & 127;     // dim
        float v = 0.0f;
        if (b < BATCH) {
            const int h = head[b];
            const int r = relation[b];
            v = emb_entity[(long long)h * EMB_DIM + d] +
                emb_relation[(long long)r * EMB_DIM + d];
        }
        sPred[idx] = v;
    }
    __syncthreads();

    if (tid < 16) {
        float s = 0.0f;
        #pragma unroll 8
        for (int d = 0; d < EMB_DIM; ++d) {
            const float v = sPred[tid * EMB_DIM + d];
            s += v * v;
        }
        sQ[tid]  = s;
        sBH[tid] = (tid < BATCH) ? bias_head[head[tid]] : 0.0f;
    }
    __syncthreads();

    const int wave  = tid >> 5;   // 0..7 : 16-entity tile per wave
    const int lane  = tid & 31;
    const int lhalf = lane >> 4;  // 0/1 -> K sub-pair selector (A & B frag layout)
    const int lmod  = lane & 15;  // A: row M ; B: col N

    const long long e0 = (long long)blockIdx.x * 128 + (long long)wave * 16;

    // A fragment source row (clamped so all 32 lanes stay active: WMMA needs EXEC==all-1s)
    long long eRow = e0 + lmod;
    if (eRow >= E) eRow = E - 1;
    const int trow = tail[eRow];

    const float* aBase = emb_entity + (long long)trow * EMB_DIM + 2 * lhalf;
    const float* bBase = sPred + lmod * EMB_DIM + 2 * lhalf;

    v8f  c  = {};
    float tn = 0.0f;   // this lane's half of ||t_row||^2

    // K = 128 as 32 chained 16x16x4 f32 WMMAs
    #pragma unroll
    for (int kc = 0; kc < EMB_DIM / 4; ++kc) {
        const v2f a  = *(const v2f*)(aBase + 4 * kc);   // global_load_b64
        const v2f bv = *(const v2f*)(bBase + 4 * kc);   // ds_load_b64
        tn += a.x * a.x + a.y * a.y;
        c = __builtin_amdgcn_wmma_f32_16x16x4_f32(
                /*neg_a=*/false, a, /*neg_b=*/false, bv,
                /*c_mod=*/(short)0, c, /*reuse_a=*/false, /*reuse_b=*/false);
    }

    // lanes l and l^16 hold the two K-halves of row (l%16): combine, park in LDS
    tn += __shfl_xor(tn, 16, 32);
    if (lane < 16) sTn[wave * 16 + lane] = tn;
    __syncthreads();

    // C/D layout: lane l, VGPR r -> (M = r + 8*(l/16), N = l%16)
    const int   b  = lmod;          // query column
    const float q  = sQ[b];
    const float bh = sBH[b];

    #pragma unroll
    for (int r = 0; r < 8; ++r) {
        const int       m = r + 8 * lhalf;
        const long long e = e0 + m;
        if (b < BATCH && e < E) {
            const int t = tail[e];
            float d2 = sTn[wave * 16 + m] - 2.0f * c[r] + q;
            d2 = fmaxf(d2, 0.0f);
            const float sc = __builtin_sqrtf(d2) + bh + bias_tail[t];
            out[(long long)b * E + e] = sc;
        }
    }
}

extern "C" void kernel_launch(void* const* d_in, const int* in_sizes, int n_in,
                              void* d_out, int out_size, void* d_ws, size_t ws_size,
                              hipStream_t stream)
{
    const int*   head         = (const int*)  d_in[0];
    const int*   relation     = (const int*)  d_in[1];
    const int*   tail         = (const int*)  d_in[2];
    const float* emb_entity   = (const float*)d_in[3];
    const float* emb_relation = (const float*)d_in[4];
    const float* bias_head    = (const float*)d_in[5];
    const float* bias_tail    = (const float*)d_in[6];
    float*       out          = (float*)      d_out;

    const int E = in_sizes[2];                 // number of candidate tails
    const int grid = (E + 127) / 128;          // 128 entities per 256-thread block

    transe_score_wmma<<<grid, 256, 0, stream>>>(head, relation, tail,
                                                emb_entity, emb_relation,
                                                bias_head, bias_tail,
                                                out, E);
}